// MovingAvgNorm_29317446763164
// MI455X (gfx1250) — compile-verified
//
#include <hip/hip_runtime.h>
#include <stdint.h>

// MovingAvgNorm: sliding-window (K=100, reflect pad) mean/std normalization over
// x of shape (B=8, T=16384, D=128) fp32. Pure bandwidth problem (~128 MiB total
// traffic -> ~5.6us floor at 23.3 TB/s). Strategy: TDM-stage each block's
// (CT + K - 1) x 128 fp32 time tile into LDS with one tensor_load_to_lds, then
// per-column running sliding sums from LDS, coalesced stores to HBM.

#define BB   8
#define TT   16384
#define DD   128
#define KK   100
#define HALF (KK / 2)            // 50
#define CT   128                 // output frames per block
#define NROWS (CT + KK - 1)      // 227 staged padded rows
#define EPSF 1e-12f

#ifndef __has_builtin
#define __has_builtin(x) 0
#endif

#if __has_builtin(__builtin_amdgcn_tensor_load_to_lds)
#define HAVE_TDM 1
#else
#define HAVE_TDM 0
#endif

typedef __attribute__((ext_vector_type(4))) unsigned int uint32x4;
typedef __attribute__((ext_vector_type(8))) int          int32x8;
typedef __attribute__((ext_vector_type(4))) int          int32x4;

__device__ __forceinline__ int reflect_t(int o) {
    o = (o < 0) ? -o : o;
    return (o >= TT) ? (2 * TT - 2 - o) : o;
}

__global__ __launch_bounds__(DD) void MovingAvgNorm_29317446763164_kernel(
    const float* __restrict__ x, float* __restrict__ out) {
    // Only __shared__ object in the kernel -> LDS byte offset 0.
    __shared__ float tile[NROWS][DD];   // 227 * 128 * 4 = 116224 B

    const int d  = threadIdx.x;         // column (0..127)
    const int b  = blockIdx.y;          // batch
    const int t0 = blockIdx.x * CT;     // first output frame of this block
    const long long base = (long long)b * TT * DD;  // element offset of batch

    const int lo = t0 - HALF;           // original time of staged row 0
    // Contiguous (non-reflected) staged rows: [r_lo, r_hi)
    int r_lo = (lo < 0) ? -lo : 0;
    int r_hi = TT - lo;
    if (r_hi > NROWS) r_hi = NROWS;

#if HAVE_TDM
    // ---- Tensor Data Mover: bulk-stage the contiguous region (wave 0 only) ----
    if (threadIdx.x < 32) {
        const unsigned int lds_addr =
            (unsigned int)((unsigned)r_lo * DD * sizeof(float)); // tile @ LDS offset 0
        const unsigned long long gaddr =
            (unsigned long long)(uintptr_t)(x + base + (long long)(lo + r_lo) * DD);
        const unsigned int nrows = (unsigned int)(r_hi - r_lo);
        const unsigned int dim0  = DD;      // elements per row
        const unsigned int str0  = DD;      // row stride in elements

        uint32x4 g0;
        g0.x = 1u;                                            // count=1, user D#
        g0.y = lds_addr;                                      // lds_addr  [63:32]
        g0.z = (unsigned int)(gaddr & 0xFFFFFFFFull);         // gaddr lo  [95:64]
        g0.w = (unsigned int)((gaddr >> 32) & 0x1FFFFFFull)   // gaddr hi  [120:96]
             | (2u << 30);                                    // type=2    [127:126]

        int32x8 g1;
        g1[0] = (int)0x00020000u;                             // mask=0, data_size=4B
        g1[1] = (int)((dim0 & 0xFFFFu) << 16);                // tensor_dim0 lo16
        g1[2] = (int)(((dim0 >> 16) & 0xFFFFu)                // tensor_dim0 hi16
                      | ((nrows & 0xFFFFu) << 16));           // tensor_dim1 lo16
        g1[3] = (int)(((nrows >> 16) & 0xFFFFu)               // tensor_dim1 hi16
                      | ((dim0 & 0xFFFFu) << 16));            // tile_dim0 = 128
        g1[4] = (int)(nrows & 0xFFFFu);                       // tile_dim1, tile_dim2=0
        g1[5] = (int)str0;                                    // tensor_dim0_stride lo32
        g1[6] = 0;                                            // stride0 hi16, stride1 lo16
        g1[7] = 0;
        int32x4 z4 = {0, 0, 0, 0};                            // groups 2/3 unused (2D)
        int32x8 z8 = {0, 0, 0, 0, 0, 0, 0, 0};
        __builtin_amdgcn_tensor_load_to_lds(g0, g1, z4, z4, z8, 0);
    }
#else
    // No TDM builtin available: stage everything via the fixup path below.
    r_lo = 0;
    r_hi = 0;
#endif

    // ---- Reflect-padded edge rows (only first/last chunk per batch) ----
    for (int r = 0; r < r_lo; ++r)
        tile[r][d] = x[base + (long long)reflect_t(lo + r) * DD + d];
    for (int r = r_hi; r < NROWS; ++r)
        tile[r][d] = x[base + (long long)reflect_t(lo + r) * DD + d];

#if HAVE_TDM
#if __has_builtin(__builtin_amdgcn_s_wait_tensorcnt)
    if (threadIdx.x < 32) __builtin_amdgcn_s_wait_tensorcnt(0);
#endif
#endif
    __syncthreads();

    // ---- Per-column sliding window from LDS ----
    float s = 0.f, s2 = 0.f;
#pragma unroll 4
    for (int j = 0; j < KK; ++j) {
        float v = tile[j][d];
        s  += v;
        s2 += v * v;
    }

    const float invK   = 1.0f / (float)KK;
    const float invKm1 = 1.0f / (float)(KK - 1);
    const long long obase = base + (long long)t0 * DD + d;

    for (int i = 0; i < CT; ++i) {
        const float xc  = tile[i + HALF][d];
        const float m   = s * invK;
        float var = (s2 - (float)KK * m * m) * invKm1;
        var = (var > 0.f) ? var : 0.f;
        const float sd = __builtin_sqrtf(var);
        out[obase + (long long)i * DD] = (xc - m) / (sd + EPSF);

        if (i + 1 < CT) {          // slide window to the next frame
            const float a = tile[i + KK][d];
            const float q = tile[i][d];
            s  += a - q;
            s2 += a * a - q * q;
        }
    }
}

extern "C" void kernel_launch(void* const* d_in, const int* in_sizes, int n_in,
                              void* d_out, int out_size, void* d_ws, size_t ws_size,
                              hipStream_t stream) {
    (void)in_sizes; (void)n_in; (void)d_ws; (void)ws_size; (void)out_size;
    const float* x = (const float*)d_in[0];
    float* out = (float*)d_out;
    dim3 grid(TT / CT, BB, 1);   // 128 x 8 = 1024 blocks
    dim3 block(DD, 1, 1);        // 128 threads = 4 wave32
    MovingAvgNorm_29317446763164_kernel<<<grid, block, 0, stream>>>(x, out);
}